// EMDLoss2D_20590073217249
// MI455X (gfx1250) — compile-verified
//
#include <hip/hip_runtime.h>
#include <math.h>

// CDNA5 / gfx1250: wave32, WMMA 16x16x4 f32
typedef __attribute__((ext_vector_type(2))) float v2f;
typedef __attribute__((ext_vector_type(8))) float v8f;

#define KDIM 64
#define BDIM 128

// ---------------------------------------------------------------------------
// WMMA tile helpers for the cumsum-as-matmul trick.
//   cdf = L * C * L^T,  L = 64x64 lower-triangular ones.
// Fragment layouts per ISA 7.12.2 (wave32):
//   A 16x4 f32 : lane m=lane&15, vgpr v, half h=lane>>4 -> A[m, 4*kb + v + 2h]
//   B 4x16 f32 : col n=lane&15,  vgpr v, half h          -> B[v + 2h, n]
//   C/D 16x16  : vgpr v, half h -> row v + 8h, col lane&15
// ---------------------------------------------------------------------------

// D-tile = C_tile(ti,:) * L^T(:,tj)   (row-wise cumsum contribution)
__device__ __forceinline__ v8f tile_cumsum_cols(const float (*C)[KDIM + 1],
                                                int ti, int tj, int m, int h)
{
    v8f acc = {};
    #pragma unroll
    for (int kb = 0; kb < 16; ++kb) {
        const int k0 = kb * 4 + 2 * h;
        v2f a, bf;
        a.x = C[ti * 16 + m][k0];
        a.y = C[ti * 16 + m][k0 + 1];
        const int n = tj * 16 + m;                 // B col index
        bf.x = (k0     <= n) ? 1.0f : 0.0f;        // L^T[k][n] = (k <= n)
        bf.y = (k0 + 1 <= n) ? 1.0f : 0.0f;
        acc = __builtin_amdgcn_wmma_f32_16x16x4_f32(
                  false, a, false, bf, (short)0, acc, false, false);
    }
    return acc;
}

// D-tile = L(ti,:) * X_tile(:,tj)     (col-wise cumsum contribution)
__device__ __forceinline__ v8f tile_cumsum_rows(const float (*X)[KDIM + 1],
                                                int ti, int tj, int m, int h)
{
    v8f acc = {};
    #pragma unroll
    for (int kb = 0; kb < 16; ++kb) {
        const int k0  = kb * 4 + 2 * h;
        const int row = ti * 16 + m;               // A row index
        v2f a, bf;
        a.x = (k0     <= row) ? 1.0f : 0.0f;       // L[row][k] = (k <= row)
        a.y = (k0 + 1 <= row) ? 1.0f : 0.0f;
        bf.x = X[k0    ][tj * 16 + m];
        bf.y = X[k0 + 1][tj * 16 + m];
        acc = __builtin_amdgcn_wmma_f32_16x16x4_f32(
                  false, a, false, bf, (short)0, acc, false, false);
    }
    return acc;
}

__device__ __forceinline__ void tile_store(float (*dst)[KDIM + 1], v8f d,
                                           int ti, int tj, int m, int h)
{
    #pragma unroll
    for (int v = 0; v < 8; ++v)
        dst[ti * 16 + v + 8 * h][tj * 16 + m] = d[v];
}

// ---------------------------------------------------------------------------
// Phase 1: per-batch 2D CDF of the counts channel via WMMA.
// One block per batch, 8 waves; each wave owns output tiles (wave) and
// (wave+8) of the 4x4 tile grid -> straight-line, uniform control flow, so
// EXEC is provably all-1s at every v_wmma (ISA requirement).
// Normalization (/(sum+eps)) commutes with the linear cumsum; the raw total
// is cdf[63][63] after the full 2D cumsum.
// ---------------------------------------------------------------------------
__global__ __launch_bounds__(256)
void emd_cdf_kernel(const float* __restrict__ hist, float* __restrict__ cdf_out)
{
    __shared__ float Cbuf[KDIM][KDIM + 1];   // counts, later final cdf (padded)
    __shared__ float Xbuf[KDIM][KDIM + 1];   // intermediate C*L^T

    const int b    = blockIdx.x;
    const int tid  = threadIdx.x;
    const int lane = tid & 31;               // wave32
    const int wave = tid >> 5;               // 0..7
    const int m    = lane & 15;
    const int h    = lane >> 4;

    const int t0 = wave,     ti0 = t0 >> 2, tj0 = t0 & 3;
    const int t1 = wave + 8, ti1 = t1 >> 2, tj1 = t1 & 3;

    const float* histB = hist    + (size_t)b * KDIM * KDIM * 3;
    float*       cdfB  = cdf_out + (size_t)b * KDIM * KDIM;

    // cooperative load of counts channel (stride-3 gather, channel 2)
    for (int e = tid; e < KDIM * KDIM; e += 256)
        Cbuf[e >> 6][e & 63] = histB[e * 3 + 2];
    __syncthreads();

    // ---- matmul 1: X = C * L^T (cumsum along axis -1) ----
    {
        v8f d0 = tile_cumsum_cols(Cbuf, ti0, tj0, m, h);
        v8f d1 = tile_cumsum_cols(Cbuf, ti1, tj1, m, h);
        tile_store(Xbuf, d0, ti0, tj0, m, h);
        tile_store(Xbuf, d1, ti1, tj1, m, h);
    }
    __syncthreads();

    // ---- matmul 2: Y = L * X (cumsum along axis -2), write into Cbuf ----
    {
        v8f d0 = tile_cumsum_rows(Xbuf, ti0, tj0, m, h);
        v8f d1 = tile_cumsum_rows(Xbuf, ti1, tj1, m, h);
        tile_store(Cbuf, d0, ti0, tj0, m, h);
        tile_store(Cbuf, d1, ti1, tj1, m, h);
    }
    __syncthreads();

    // total raw count = full 2D cumsum at [63][63]; scale and store
    const float scale = 1.0f / (Cbuf[KDIM - 1][KDIM - 1] + 1e-6f);
    for (int e = tid; e < KDIM * KDIM; e += 256)
        cdfB[e] = Cbuf[e >> 6][e & 63] * scale;
}

// ---------------------------------------------------------------------------
// Phase 2: one block per (b, j) pair; 64x64 (i, l) tile.
//   acc += sqrt((px[b,i,j]-tx[b,j,l])^2 + (py[b,i,j]-ty[b,j,l])^2)
//          * |cdfP[b,i,j] - cdfT[b,j,l]|
// Thread t: l = t&63 fixed (teacher side in registers), i = (t>>6)*16..+15
// (pred side broadcast from LDS). Deterministic block reduction -> partial.
// ---------------------------------------------------------------------------
__global__ __launch_bounds__(256)
void emd_pair_kernel(const float* __restrict__ student,
                     const float* __restrict__ teacher,
                     const float* __restrict__ cdfP,
                     const float* __restrict__ cdfT,
                     float* __restrict__ partial)
{
    __shared__ float spx[KDIM], spy[KDIM], scp[KDIM];
    __shared__ float wsum[8];

    const int bid = blockIdx.x;
    const int b   = bid >> 6;
    const int j   = bid & 63;
    const int tid = threadIdx.x;

    if (tid < KDIM) {                         // pred side: strided gather into LDS
        const int i = tid;
        const size_t off = ((size_t)(b * KDIM + i) * KDIM + j) * 3;
        spx[i] = student[off + 0];
        spy[i] = student[off + 1];
        scp[i] = cdfP[(b * KDIM + i) * KDIM + j];
    }
    const int l  = tid & 63;
    const int i0 = tid >> 6;
    const size_t toff = ((size_t)(b * KDIM + j) * KDIM + l) * 3;
    const float tx = teacher[toff + 0];
    const float ty = teacher[toff + 1];
    const float ct = cdfT[(b * KDIM + j) * KDIM + l];
    __syncthreads();

    float acc = 0.0f;
    #pragma unroll
    for (int ii = 0; ii < 16; ++ii) {
        const int i = i0 * 16 + ii;
        const float dx = spx[i] - tx;
        const float dy = spy[i] - ty;
        const float g  = sqrtf(dx * dx + dy * dy);
        acc += g * fabsf(scp[i] - ct);
    }

    // wave32 shuffle reduction, then 8 wave-partials in fixed order
    #pragma unroll
    for (int o = 16; o > 0; o >>= 1)
        acc += __shfl_down(acc, o, 32);
    if ((tid & 31) == 0) wsum[tid >> 5] = acc;
    __syncthreads();
    if (tid == 0) {
        float s = 0.0f;
        #pragma unroll
        for (int w = 0; w < 8; ++w) s += wsum[w];
        partial[bid] = s;
    }
}

// ---------------------------------------------------------------------------
// Phase 3: deterministic reduction of the 8192 partials -> mean over (b, i).
// ---------------------------------------------------------------------------
__global__ __launch_bounds__(256)
void emd_final_kernel(const float* __restrict__ partial, float* __restrict__ out)
{
    __shared__ float wsum[8];
    const int tid = threadIdx.x;
    float acc = 0.0f;
    for (int e = tid; e < BDIM * KDIM; e += 256) acc += partial[e];
    #pragma unroll
    for (int o = 16; o > 0; o >>= 1)
        acc += __shfl_down(acc, o, 32);
    if ((tid & 31) == 0) wsum[tid >> 5] = acc;
    __syncthreads();
    if (tid == 0) {
        float s = 0.0f;
        #pragma unroll
        for (int w = 0; w < 8; ++w) s += wsum[w];
        out[0] = s * (1.0f / (float)(BDIM * KDIM));   // jnp.mean over [B, K]
    }
}

extern "C" void kernel_launch(void* const* d_in, const int* in_sizes, int n_in,
                              void* d_out, int out_size, void* d_ws, size_t ws_size,
                              hipStream_t stream)
{
    const float* student = (const float*)d_in[0];   // [B, K, K, 3] f32
    const float* teacher = (const float*)d_in[1];   // [B, K, K, 3] f32

    // workspace: cdfP [B*K*K] | cdfT [B*K*K] | partials [B*K]  (~4.23 MB)
    float* cdfP    = (float*)d_ws;
    float* cdfT    = cdfP + (size_t)BDIM * KDIM * KDIM;
    float* partial = cdfT + (size_t)BDIM * KDIM * KDIM;

    emd_cdf_kernel<<<BDIM, 256, 0, stream>>>(student, cdfP);
    emd_cdf_kernel<<<BDIM, 256, 0, stream>>>(teacher, cdfT);
    emd_pair_kernel<<<BDIM * KDIM, 256, 0, stream>>>(student, teacher, cdfP, cdfT, partial);
    emd_final_kernel<<<1, 256, 0, stream>>>(partial, (float*)d_out);
}